// DCNNv2_36112085025447
// MI455X (gfx1250) — compile-verified
//
#include <hip/hip_runtime.h>
#include <math.h>

typedef float v2f __attribute__((ext_vector_type(2)));
typedef float v8f __attribute__((ext_vector_type(8)));

#define D        128
#define KNODES   64
#define DEG_IN_  8
#define DEG_EXT_ 16
#define NGRAPH   2000
#define NBATCH   1024
// LDS A tile: K-major [256][64] floats = 64KB exactly.
// Fragment read sA[kb*64 + l] -> bank = l, conflict-free.
#define AK       256
#define AM       64

__device__ __forceinline__ v8f wmma4(v2f a, v2f b, v8f c) {
  // V_WMMA_F32_16X16X4_F32: D = A(16x4) * B(4x16) + C, f32 throughout
  return __builtin_amdgcn_wmma_f32_16x16x4_f32(false, a, false, b, (short)0, c,
                                               false, false);
}

// Shared WMMA GEMM core: (64 x 256) @ (256 x 128).
// sA: LDS, K-major [256][64].  Bt: global, row-major [256][128] (Bt[k][n]).
// Wave handles N column [n0, n0+16) and all four 16-row M tiles.
__device__ __forceinline__ void gemm_tiles(const float* sA, const float* Bt,
                                           int lane, int n0, v8f acc[4]) {
  const int half = lane >> 4;       // 0: K sub 0/1, 1: K sub 2/3
  const int l    = lane & 15;       // M row (A) / N col (B) within tile
  const int n    = n0 + l;
#pragma unroll 2
  for (int k = 0; k < AK; k += 4) {
    const int kb = k + half * 2;
    v2f b;
    b.x = Bt[kb * D + n];
    b.y = Bt[kb * D + D + n];
#pragma unroll
    for (int mt = 0; mt < 4; ++mt) {
      v2f a;
      a.x = sA[kb * AM + mt * 16 + l];
      a.y = sA[kb * AM + AM + mt * 16 + l];
      acc[mt] = wmma4(a, b, acc[mt]);
    }
  }
}

// ---------------- K0: pre-transpose the weight matrices ----------------
// Bt[0]: [W|M]^T (256x128)  Bt[1]: [U|V]^T  Bt[2]: W1^T (W1 is 128x256)
__global__ void prep_bt(const float* __restrict__ W, const float* __restrict__ M,
                        const float* __restrict__ U, const float* __restrict__ V,
                        const float* __restrict__ W1, float* __restrict__ Bt) {
  int id = blockIdx.x * blockDim.x + threadIdx.x;
  if (id >= 3 * AK * D) return;
  int table = id / (AK * D);
  int rem   = id % (AK * D);
  int k = rem / D, n = rem % D;
  float v;
  if (table == 0)      v = (k < D) ? W[n * D + k] : M[n * D + (k - D)];
  else if (table == 1) v = (k < D) ? U[n * D + k] : V[n * D + (k - D)];
  else                 v = W1[n * (2 * D) + k];
  Bt[id] = v;
}

// ---------------- K1: gather + (64x256)@(256x128) + relu-colsum ----------------
__global__ void __launch_bounds__(256)
k1_aggregate_gemm(const int* __restrict__ node_type,
                  const int* __restrict__ nbr_type,
                  const float* __restrict__ impact,
                  const float* __restrict__ Bt1,
                  float* __restrict__ per_graph) {
  __shared__ float sA[AK * AM];  // 64 KB, K-major
  const int g = blockIdx.x;
  const int t = threadIdx.x;
  // Phase 1: gather. thread -> (row = t&63, 32-col slice = (t>>6)*32)
  {
    const int row   = t & 63;
    const int cbase = (t >> 6) * 32;
    const int sidx  = node_type[g * KNODES + row];
    const float* src = impact + sidx * D + cbase;
#pragma unroll
    for (int i = 0; i < 32; i += 4) {
      float4 v = *(const float4*)(src + i);
      sA[(cbase + i + 0) * AM + row] = v.x;
      sA[(cbase + i + 1) * AM + row] = v.y;
      sA[(cbase + i + 2) * AM + row] = v.z;
      sA[(cbase + i + 3) * AM + row] = v.w;
    }
    float4 acc4[8];
#pragma unroll
    for (int i = 0; i < 8; ++i) acc4[i] = make_float4(0.f, 0.f, 0.f, 0.f);
    const int* nb = nbr_type + (g * KNODES + row) * DEG_IN_;
    for (int d = 0; d < DEG_IN_; ++d) {
      const float* s = impact + nb[d] * D + cbase;
#pragma unroll
      for (int i = 0; i < 8; ++i) {
        float4 v = *(const float4*)(s + i * 4);
        acc4[i].x += v.x; acc4[i].y += v.y; acc4[i].z += v.z; acc4[i].w += v.w;
      }
    }
#pragma unroll
    for (int i = 0; i < 8; ++i) {
      sA[(D + cbase + i * 4 + 0) * AM + row] = acc4[i].x;
      sA[(D + cbase + i * 4 + 1) * AM + row] = acc4[i].y;
      sA[(D + cbase + i * 4 + 2) * AM + row] = acc4[i].z;
      sA[(D + cbase + i * 4 + 3) * AM + row] = acc4[i].w;
    }
  }
  __syncthreads();
  // Phase 2: WMMA GEMM + relu + column sum over the 64 node rows
  const int lane = t & 31;
  const int n0   = (t >> 5) * 16;
  v8f acc[4];
#pragma unroll
  for (int i = 0; i < 4; ++i) acc[i] = (v8f){};
  gemm_tiles(sA, Bt1, lane, n0, acc);
  float partial = 0.f;
#pragma unroll
  for (int mt = 0; mt < 4; ++mt)
#pragma unroll
    for (int r = 0; r < 8; ++r) partial += fmaxf(acc[mt][r], 0.f);
  partial += __shfl_xor(partial, 16);   // combine M halves (lane <-> lane+16)
  if (lane < 16) per_graph[g * D + n0 + lane] = partial;
}

// ---------------- softmax over a 128-wide row (one wave per row) ----------------
__global__ void k_softmax(const float* __restrict__ in, float* __restrict__ out,
                          int do_relu) {
  const int g = blockIdx.x;
  const int lane = threadIdx.x;  // 32 lanes, 4 cols each
  float v[4];
#pragma unroll
  for (int i = 0; i < 4; ++i) {
    float x = in[g * D + lane + 32 * i];
    if (do_relu) x = fmaxf(x, 0.f);
    v[i] = x;
  }
  float m = fmaxf(fmaxf(v[0], v[1]), fmaxf(v[2], v[3]));
  for (int off = 16; off > 0; off >>= 1) m = fmaxf(m, __shfl_xor(m, off));
  float s = 0.f;
#pragma unroll
  for (int i = 0; i < 4; ++i) { v[i] = __expf(v[i] - m); s += v[i]; }
  for (int off = 16; off > 0; off >>= 1) s += __shfl_xor(s, off);
  const float inv = 1.f / s;
#pragma unroll
  for (int i = 0; i < 4; ++i) out[g * D + lane + 32 * i] = v[i] * inv;
}

// ---------------- K3: [E | sum E[ext_nbr]] @ [U|V]^T -> ext ----------------
__global__ void __launch_bounds__(256)
k3_ext_gemm(const int* __restrict__ ext_nbr, const float* __restrict__ E,
            const float* __restrict__ Bt2, float* __restrict__ ext) {
  __shared__ float sA[AK * AM];
  const int g0 = blockIdx.x * 64;
  const int t  = threadIdx.x;
  {
    const int row = t & 63;
    const int cbase = (t >> 6) * 32;
    const int gg = g0 + row;
    if (gg < NGRAPH) {
      const float* src = E + gg * D + cbase;
#pragma unroll
      for (int i = 0; i < 32; i += 4) {
        float4 v = *(const float4*)(src + i);
        sA[(cbase + i + 0) * AM + row] = v.x;
        sA[(cbase + i + 1) * AM + row] = v.y;
        sA[(cbase + i + 2) * AM + row] = v.z;
        sA[(cbase + i + 3) * AM + row] = v.w;
      }
      float4 acc4[8];
#pragma unroll
      for (int i = 0; i < 8; ++i) acc4[i] = make_float4(0.f, 0.f, 0.f, 0.f);
      const int* nb = ext_nbr + gg * DEG_EXT_;
      for (int j = 0; j < DEG_EXT_; ++j) {
        const float* s = E + nb[j] * D + cbase;
#pragma unroll
        for (int i = 0; i < 8; ++i) {
          float4 v = *(const float4*)(s + i * 4);
          acc4[i].x += v.x; acc4[i].y += v.y; acc4[i].z += v.z; acc4[i].w += v.w;
        }
      }
#pragma unroll
      for (int i = 0; i < 8; ++i) {
        sA[(D + cbase + i * 4 + 0) * AM + row] = acc4[i].x;
        sA[(D + cbase + i * 4 + 1) * AM + row] = acc4[i].y;
        sA[(D + cbase + i * 4 + 2) * AM + row] = acc4[i].z;
        sA[(D + cbase + i * 4 + 3) * AM + row] = acc4[i].w;
      }
    } else {
#pragma unroll
      for (int i = 0; i < 32; ++i) {
        sA[(cbase + i) * AM + row]     = 0.f;
        sA[(D + cbase + i) * AM + row] = 0.f;
      }
    }
  }
  __syncthreads();
  const int lane = t & 31;
  const int n0   = (t >> 5) * 16;
  v8f acc[4];
#pragma unroll
  for (int i = 0; i < 4; ++i) acc[i] = (v8f){};
  gemm_tiles(sA, Bt2, lane, n0, acc);
  const int half = lane >> 4, l = lane & 15, n = n0 + l;
#pragma unroll
  for (int mt = 0; mt < 4; ++mt)
#pragma unroll
    for (int r = 0; r < 8; ++r) {
      const int gg = g0 + mt * 16 + half * 8 + r;
      if (gg < NGRAPH) ext[gg * D + n] = acc[mt][r];
    }
}

// ---------------- K5: pair features -> MLP head -> softmax(2) ----------------
__global__ void __launch_bounds__(256)
k5_head(const int* __restrict__ batch, const float* __restrict__ X,
        const float* __restrict__ Bt3, const float* __restrict__ b1,
        const float* __restrict__ W2, const float* __restrict__ b2,
        float* __restrict__ out) {
  __shared__ float sA[AK * AM];  // reused as h[64][128] after the GEMM
  const int b0 = blockIdx.x * 64;
  const int t  = threadIdx.x;
  {
    const int row = t & 63;
    const int cbase = (t >> 6) * 32;
    const int rb = b0 + row;
    const int i1 = batch[rb * 2 + 0];
    const int i2 = batch[rb * 2 + 1];
    const float* p1 = X + i1 * D + cbase;
    const float* p2 = X + i2 * D + cbase;
#pragma unroll
    for (int i = 0; i < 32; ++i) {
      const float a = p1[i], b = p2[i];
      sA[(cbase + i) * AM + row]     = a * b;  // e1*e2
      sA[(D + cbase + i) * AM + row] = a + b;  // e1+e2
    }
  }
  __syncthreads();
  const int lane = t & 31;
  const int n0   = (t >> 5) * 16;
  v8f acc[4];
#pragma unroll
  for (int i = 0; i < 4; ++i) acc[i] = (v8f){};
  gemm_tiles(sA, Bt3, lane, n0, acc);
  __syncthreads();  // all reads of sA done; reuse it for h
  float* sH = sA;   // [64][128]
  const int half = lane >> 4, l = lane & 15, n = n0 + l;
  const float bias = b1[n];
#pragma unroll
  for (int mt = 0; mt < 4; ++mt)
#pragma unroll
    for (int r = 0; r < 8; ++r)
      sH[(mt * 16 + half * 8 + r) * D + n] = fmaxf(acc[mt][r] + bias, 0.f);
  __syncthreads();
  if (t < 64) {
    const int rb = b0 + t;
    const float* h = sH + t * D;
    float z0 = b2[0], z1 = b2[1];
    for (int k = 0; k < D; ++k) {
      const float hv = h[k];
      z0 += hv * W2[k];
      z1 += hv * W2[D + k];
    }
    const float m  = fmaxf(z0, z1);
    const float e0 = __expf(z0 - m), e1 = __expf(z1 - m);
    const float inv = 1.f / (e0 + e1);
    out[rb * 2 + 0] = e0 * inv;
    out[rb * 2 + 1] = e1 * inv;
  }
}

extern "C" void kernel_launch(void* const* d_in, const int* in_sizes, int n_in,
                              void* d_out, int out_size, void* d_ws, size_t ws_size,
                              hipStream_t stream) {
  const int*   batch     = (const int*)d_in[0];
  const int*   node_type = (const int*)d_in[1];
  const int*   nbr_type  = (const int*)d_in[2];
  const int*   ext_nbr   = (const int*)d_in[3];
  const float* impact    = (const float*)d_in[4];
  const float* W         = (const float*)d_in[5];
  const float* M         = (const float*)d_in[6];
  const float* U         = (const float*)d_in[7];
  const float* V         = (const float*)d_in[8];
  const float* W1        = (const float*)d_in[9];
  const float* b1        = (const float*)d_in[10];
  const float* W2        = (const float*)d_in[11];
  const float* b2        = (const float*)d_in[12];
  float* ws = (float*)d_ws;
  float* Bt1 = ws;                 // 256*128
  float* Bt2 = ws + 32768;         // 256*128
  float* Bt3 = ws + 65536;         // 256*128
  float* per_graph = ws + 98304;   // 2000*128
  float* E   = per_graph + 256000; // 2000*128
  float* ext = E + 256000;         // 2000*128
  float* X   = ext + 256000;       // 2000*128
  float* out = (float*)d_out;      // 1024*2

  prep_bt<<<384, 256, 0, stream>>>(W, M, U, V, W1, ws);
  k1_aggregate_gemm<<<NGRAPH, 256, 0, stream>>>(node_type, nbr_type, impact,
                                                Bt1, per_graph);
  k_softmax<<<NGRAPH, 32, 0, stream>>>(per_graph, E, 0);
  k3_ext_gemm<<<32, 256, 0, stream>>>(ext_nbr, E, Bt2, ext);
  k_softmax<<<NGRAPH, 32, 0, stream>>>(ext, X, 1);
  k5_head<<<16, 256, 0, stream>>>(batch, X, Bt3, b1, W2, b2, out);
}